// SimCLR_Loss_3049426780441
// MI455X (gfx1250) — compile-verified
//
#include <hip/hip_runtime.h>
#include <math.h>

typedef float v2f __attribute__((ext_vector_type(2)));
typedef float v8f __attribute__((ext_vector_type(8)));

#define N2    8192      // 2*4096 rows
#define BHALF 4096
#define DIM   256
#define KSTEPS (DIM / 4)      // 64 k-steps of K=4
#define JSPLIT 8              // J-chunks per row-tile
#define JTILES_PER_CHUNK 64   // (8192/16)/JSPLIT

// ---------------- Kernel 1: row-normalize z -> e ----------------
__global__ __launch_bounds__(256) void k_normalize(const float* __restrict__ z,
                                                   float* __restrict__ e) {
    int row = blockIdx.x;            // 0..8191
    int col = threadIdx.x;           // 0..255
    // e row r: r<4096 -> z[r,0,:], else z[r-4096,1,:]
    int zrow = (row < BHALF) ? (row * 2) : ((row - BHALF) * 2 + 1);
    float v = z[zrow * DIM + col];
    float s = v * v;
#pragma unroll
    for (int m = 16; m >= 1; m >>= 1) s += __shfl_xor(s, m);
    __shared__ float red[8];
    if ((threadIdx.x & 31) == 0) red[threadIdx.x >> 5] = s;
    __syncthreads();
    float tot = 0.0f;
#pragma unroll
    for (int i = 0; i < 8; ++i) tot += red[i];
    e[row * DIM + col] = v / sqrtf(tot);
}

// ---------------- Per-tile epilogue: exp + masked accumulation ----------------
__device__ __forceinline__ void tile_tail(v8f c, int i0, int half, int j,
                                          v8f& negacc, float* __restrict__ out_pos) {
    // D layout: lanes 0-15 -> (m=v, n=lane); lanes 16-31 -> (m=v+8, n=lane-16)
#pragma unroll
    for (int v = 0; v < 8; ++v) {
        const int i = i0 + v + 8 * half;
        const float val = __expf(c[v] * 10.0f);   // exp(dot / TAU)
        if (((j ^ i) & (BHALF - 1)) != 0)         // j%4096 != i%4096
            negacc[v] += val;
        if (j == ((i + BHALF) & (N2 - 1)))        // the positive partner
            out_pos[i] = val;                     // unique writer per i
    }
}

// ---------------- Kernel 2: fused Gram + exp + masked row sums ----------------
// Block: 8 waves, all sharing one jc chunk (B tiles hit WGP cache),
// wave handles I-tile = (blockIdx>>3)*8 + wave. A K-slice hoisted to VGPRs.
// Two J-tiles in flight per wave -> two independent f32 WMMA chains.
__global__ __launch_bounds__(256) void k_gram(const float* __restrict__ e,
                                              float* __restrict__ negpart,
                                              float* __restrict__ out_pos) {
    const int lane = threadIdx.x & 31;
    const int wave = threadIdx.x >> 5;
    const int jc     = blockIdx.x & 7;                 // shared by all 8 waves
    const int tile_i = (blockIdx.x >> 3) * 8 + wave;   // 0..511
    const int i0     = tile_i * 16;

    const int half = lane >> 4;         // 0 or 1
    const int r    = lane & 15;         // row-in-tile / col-in-tile

    // Hoist this wave's entire A K-slice into registers (64 x v2f = 128 VGPRs).
    const float* arow = e + (size_t)(i0 + r) * DIM + 2 * half;
    v2f afrag[KSTEPS];
#pragma unroll
    for (int t = 0; t < KSTEPS; ++t) afrag[t] = *(const v2f*)(arow + 4 * t);

    v8f negacc = {};
    const int jbase = jc * (JTILES_PER_CHUNK * 16);

    for (int jt = 0; jt < JTILES_PER_CHUNK; jt += 2) {
        const int j0 = jbase + jt * 16;
        const float* brow0 = e + (size_t)(j0 + r) * DIM + 2 * half;
        const float* brow1 = brow0 + 16 * DIM;

        v8f c0 = {};
        v8f c1 = {};
#pragma unroll
        for (int t = 0; t < KSTEPS; ++t) {
            v2f b0 = *(const v2f*)(brow0 + 4 * t);
            v2f b1 = *(const v2f*)(brow1 + 4 * t);
            c0 = __builtin_amdgcn_wmma_f32_16x16x4_f32(
                     false, afrag[t], false, b0, (short)0, c0, false, false);
            c1 = __builtin_amdgcn_wmma_f32_16x16x4_f32(
                     false, afrag[t], false, b1, (short)0, c1, false, false);
        }

        tile_tail(c0, i0, half, j0 + r,      negacc, out_pos);
        tile_tail(c1, i0, half, j0 + 16 + r, negacc, out_pos);
    }

    // Reduce neg across the 16 lanes sharing each row, store per-chunk partial.
#pragma unroll
    for (int v = 0; v < 8; ++v) {
        float s = negacc[v];
        s += __shfl_xor(s, 1);
        s += __shfl_xor(s, 2);
        s += __shfl_xor(s, 4);
        s += __shfl_xor(s, 8);
        if (r == 0)
            negpart[jc * N2 + i0 + v + 8 * half] = s;
    }
}

// ---------------- Kernel 3: combine partials, per-row loss terms ----------------
__global__ __launch_bounds__(256) void k_finalize(const float* __restrict__ negpart,
                                                  const float* __restrict__ pos,
                                                  float* __restrict__ neg_out,
                                                  float* __restrict__ loss_part) {
    int i = blockIdx.x * 256 + threadIdx.x;   // 0..8191
    float n = 0.0f;
#pragma unroll
    for (int jc = 0; jc < JSPLIT; ++jc) n += negpart[jc * N2 + i];
    neg_out[i] = n;
    float p = pos[i];
    float term = logf(p / (p + n));
#pragma unroll
    for (int m = 16; m >= 1; m >>= 1) term += __shfl_xor(term, m);
    __shared__ float red[8];
    if ((threadIdx.x & 31) == 0) red[threadIdx.x >> 5] = term;
    __syncthreads();
    if (threadIdx.x == 0) {
        float t = 0.0f;
#pragma unroll
        for (int k = 0; k < 8; ++k) t += red[k];
        loss_part[blockIdx.x] = t;
    }
}

// ---------------- Kernel 4: final scalar loss ----------------
__global__ __launch_bounds__(32) void k_loss(const float* __restrict__ loss_part,
                                             float* __restrict__ loss) {
    float t = loss_part[threadIdx.x];   // 32 partials, one wave
#pragma unroll
    for (int m = 16; m >= 1; m >>= 1) t += __shfl_xor(t, m);
    if (threadIdx.x == 0) *loss = -t / (float)N2;
}

extern "C" void kernel_launch(void* const* d_in, const int* in_sizes, int n_in,
                              void* d_out, int out_size, void* d_ws, size_t ws_size,
                              hipStream_t stream) {
    const float* z = (const float*)d_in[0];      // (4096, 2, 256) f32
    float* out = (float*)d_out;                  // [loss, pos(8192), neg(8192)]

    float* e        = (float*)d_ws;              // 8192*256 f32  (8 MB)
    float* negpart  = e + (size_t)N2 * DIM;      // 8*8192 f32    (256 KB)
    float* losspart = negpart + (size_t)JSPLIT * N2;  // 32 f32

    float* out_loss = out;
    float* out_pos  = out + 1;
    float* out_neg  = out + 1 + N2;

    k_normalize<<<N2, 256, 0, stream>>>(z, e);
    k_gram<<<(N2 / 16) * JSPLIT / 8, 256, 0, stream>>>(e, negpart, out_pos);
    k_finalize<<<N2 / 256, 256, 0, stream>>>(negpart, out_pos, out_neg, losspart);
    k_loss<<<1, 32, 0, stream>>>(losspart, out_loss);
}